// ConvBin_3100966388327
// MI455X (gfx1250) — compile-verified
//
#include <hip/hip_runtime.h>

typedef __attribute__((ext_vector_type(2))) float v2f;
typedef __attribute__((ext_vector_type(4))) float v4f;
typedef __attribute__((ext_vector_type(8))) float v8f;

#define KDIM 512
#define NDIM 32
#define JSTEPS (KDIM / 8)          // 64 outer steps; each = 2 WMMA k-steps
#define WAVES_PER_BLOCK 8
#define BLOCK_THREADS 256
#define ROWS_PER_BLOCK (WAVES_PER_BLOCK * 16)   // 128 rows of M per block

// GEMM: out[M,32] = sign( x[M,512] @ W[512,32] ), M = 32*64*64.
// fp32 WMMA (V_WMMA_F32_16X16X4_F32). K is software-permuted so each lane's
// A data for two consecutive WMMA steps is one contiguous global_load_b128;
// W is staged in LDS pre-permuted into the matching B-fragment order.
//
// K permutation (j = 0..63, s = 0..1):
//   step 2j+s, hw K0..K1 (lanes 0-15)  -> K = 8j + 2s + {0,1}
//   step 2j+s, hw K2..K3 (lanes 16-31) -> K = 8j + 2s + 4 + {0,1}
// => lane<16 holds x[row, 8j..8j+3], lane>=16 holds x[row, 8j+4..8j+7].
__global__ __launch_bounds__(BLOCK_THREADS) void
conv1x1_sign_wmma(const float* __restrict__ x, const float* __restrict__ w,
                  float* __restrict__ out, int mrows)
{
    // B fragments: [j(64)][s(2)][ntile(2)][lane(32)] -> float2 {B[k0][n], B[k0+1][n]}
    // with k0 = 8j + 2s + 4*(lane>=16), n = ntile*16 + (lane&15).   64 KB.
    __shared__ v2f lds_b[JSTEPS * 128];

    for (int idx = threadIdx.x; idx < JSTEPS * 128; idx += BLOCK_THREADS) {
        const int lane_i = idx & 31;
        const int tile   = (idx >> 5) & 1;
        const int s      = (idx >> 6) & 1;
        const int j      = idx >> 7;
        const int k0 = 8 * j + 2 * s + ((lane_i >> 4) << 2);
        const int n  = tile * 16 + (lane_i & 15);
        v2f v;
        v.x = w[k0 * NDIM + n];
        v.y = w[(k0 + 1) * NDIM + n];
        lds_b[idx] = v;
    }
    __syncthreads();

    const int wave = threadIdx.x >> 5;
    const int lane = threadIdx.x & 31;
    const long mbase = ((long)blockIdx.x * WAVES_PER_BLOCK + wave) * 16;
    if (mbase >= mrows) return;            // wave-uniform: EXEC stays all-1s

    const int  mrow  = lane & 15;
    const int  khalf = (lane >> 4) << 2;   // 0 or 4
    const float* arow = x + (mbase + mrow) * (long)KDIM + khalf;
    const v2f*   bptr = lds_b + lane;

    v8f acc0 = {};   // C tile, N = 0..15
    v8f acc1 = {};   // C tile, N = 16..31

#pragma unroll 2
    for (int j = 0; j < JSTEPS; ++j) {
        v4f a = *(const v4f*)(arow + 8 * j);     // 16B global load, 16B-aligned
        v2f a0 = { a.x, a.y };                   // step s=0 operand (regs +0,+1)
        v2f a1 = { a.z, a.w };                   // step s=1 operand (regs +2,+3)
        v2f b00 = bptr[j * 128];                 // s=0, N-tile 0
        v2f b01 = bptr[j * 128 + 32];            // s=0, N-tile 1
        v2f b10 = bptr[j * 128 + 64];            // s=1, N-tile 0
        v2f b11 = bptr[j * 128 + 96];            // s=1, N-tile 1
        acc0 = __builtin_amdgcn_wmma_f32_16x16x4_f32(
                   false, a0, false, b00, (short)0, acc0, false, false);
        acc1 = __builtin_amdgcn_wmma_f32_16x16x4_f32(
                   false, a0, false, b01, (short)0, acc1, false, false);
        acc0 = __builtin_amdgcn_wmma_f32_16x16x4_f32(
                   false, a1, false, b10, (short)0, acc0, false, false);
        acc1 = __builtin_amdgcn_wmma_f32_16x16x4_f32(
                   false, a1, false, b11, (short)0, acc1, false, false);
    }

    // C/D layout: VGPR i, lanes 0-15 -> M=i, lanes 16-31 -> M=i+8; N = lane&15.
    const int nlo = lane & 15;
    const int mhi = (lane >> 4) << 3;
    float* orow = out + (mbase + mhi) * (long)NDIM + nlo;
#pragma unroll
    for (int i = 0; i < 8; ++i) {
        const float c0 = acc0[i];
        const float c1 = acc1[i];
        // forward value of t + sg(sign(tanh(y)) - t) is sign(tanh(y)) == sign(y)
        orow[i * NDIM]      = (c0 > 0.f) ? 1.f : ((c0 < 0.f) ? -1.f : 0.f);
        orow[i * NDIM + 16] = (c1 > 0.f) ? 1.f : ((c1 < 0.f) ? -1.f : 0.f);
    }
}

extern "C" void kernel_launch(void* const* d_in, const int* in_sizes, int n_in,
                              void* d_out, int out_size, void* d_ws, size_t ws_size,
                              hipStream_t stream) {
    const float* x = (const float*)d_in[0];   // [32,64,64,512] fp32, NHWC
    const float* w = (const float*)d_in[1];   // [1,1,512,32]  fp32, HWIO
    float* out = (float*)d_out;               // [32,64,64,32] fp32

    const int mrows  = in_sizes[0] / KDIM;    // 131072
    const int blocks = (mrows + ROWS_PER_BLOCK - 1) / ROWS_PER_BLOCK;  // 1024
    conv1x1_sign_wmma<<<blocks, BLOCK_THREADS, 0, stream>>>(x, w, out, mrows);
}